// RmiModel_63788854280943
// MI455X (gfx1250) — compile-verified
//
#include <hip/hip_runtime.h>
#include <math.h>

// ---------------- problem geometry ----------------
#define BATCH   32768
#define NCH     7        // t + 3 gyro + 3 accel channels
#define NT      200      // timesteps
#define BQ      64       // batches per block (one thread per batch, 2 waves)
#define TN      16       // timesteps per LDS tile
#define ROWS    (BQ * NCH)          // 448 tensor rows per block tile
#define ROWF    (TN + 1)            // 17 floats per LDS row (TDM pad: +1 DWORD / 16 DWORDs)
#define BUF_FLOATS (ROWS * ROWF)    // 7616 floats = 30464 B per buffer
#define NTILES  ((NT + TN - 1) / TN)  // 13

typedef __attribute__((ext_vector_type(4))) unsigned int v4u;
typedef __attribute__((ext_vector_type(8))) int          v8i;
typedef __attribute__((ext_vector_type(4))) int          v4i;

// Issue one TDM 2D tile load: 448 rows x 16 floats, row stride 200 floats,
// into LDS with 1-DWORD padding after every 16 DWORDs (17-float LDS rows).
// Descriptor packing per CDNA5 ISA.
// clang-23 toolchain: 6-arg builtin (v4u, v8i, v4i, v4i, v8i, i32 cpol).
__device__ __forceinline__ void tdm_issue(unsigned ldsAddr, unsigned long long gAddr) {
    v4u g0;
    g0.x = 1u;                                            // count=1 (valid user descriptor)
    g0.y = ldsAddr;                                       // LDS byte address
    g0.z = (unsigned)(gAddr & 0xffffffffull);             // global_addr[31:0]
    g0.w = (unsigned)((gAddr >> 32) & 0x1ffffffull)       // global_addr[56:32]
         | (2u << 30);                                    // type=2 ("image")
    v8i g1;
    g1[0] = (int)((2u << 16)      // data_size = 4 bytes
                | (1u << 20)      // pad_enable
                | (3u << 22)      // pad_interval code 3 = every 16 DWORDs
                | (0u << 25));    // pad_amount  code 0 = 1 DWORD
    g1[1] = (int)((unsigned)NT   << 16);                  // tensor_dim0 lo16 (=200)
    g1[2] = (int)(((unsigned)NT >> 16) | ((unsigned)ROWS << 16)); // dim0 hi16 | tensor_dim1 lo16
    g1[3] = (int)(((unsigned)ROWS >> 16) | ((unsigned)TN << 16)); // dim1 hi16 | tile_dim0
    g1[4] = (int)((unsigned)ROWS);                        // tile_dim1 (=448), tile_dim2 = 0
    g1[5] = (int)((unsigned)NT);                          // tensor_dim0_stride[31:0] = 200
    g1[6] = (int)(((unsigned)NT & 0xffffu) << 16);        // stride0 hi16=0 | tensor_dim1_stride lo16=200
    g1[7] = 0;                                            // tensor_dim1_stride[47:16] = 0
    v4i z4 = {0, 0, 0, 0};                                // groups 2/3 unused (2D tensor)
    v8i z8 = {0, 0, 0, 0, 0, 0, 0, 0};                    // trailing group (unused dims)
    __builtin_amdgcn_tensor_load_to_lds(g0, g1, z4, z4, z8, 0);
}

__global__ __launch_bounds__(BQ)
void rmi_preint_kernel(const float* __restrict__ x,      // (B, 7, 200)
                       const float* __restrict__ calib,  // (6, 6)
                       const float* __restrict__ bias6,  // (6, 1)
                       float* __restrict__ out)          // (B, 15)
{
    __shared__ float lds[2 * BUF_FLOATS];                 // 60,928 B, double buffered

    const int tid = threadIdx.x;
    const int b   = blockIdx.x * BQ + tid;

    // Calibration matrix M = I + calib and bias: uniform -> scalar loads, live in regs.
    float M[6][6], bv[6];
#pragma unroll
    for (int i = 0; i < 6; ++i) {
#pragma unroll
        for (int j = 0; j < 6; ++j)
            M[i][j] = calib[i * 6 + j] + (i == j ? 1.0f : 0.0f);
        bv[i] = bias6[i];
    }

    // Integration state
    float DC00 = 1.f, DC01 = 0.f, DC02 = 0.f;
    float DC10 = 0.f, DC11 = 1.f, DC12 = 0.f;
    float DC20 = 0.f, DC21 = 0.f, DC22 = 1.f;
    float DV0 = 0.f, DV1 = 0.f, DV2 = 0.f;
    float DR0 = 0.f, DR1 = 0.f, DR2 = 0.f;

    // carried previous sample (step n uses imu[n] and dt = t[n+1]-t[n])
    float tP = 0.f, wP0 = 0.f, wP1 = 0.f, wP2 = 0.f, aP0 = 0.f, aP1 = 0.f, aP2 = 0.f;
    bool have = false;

    const unsigned long long gbase =
        (unsigned long long)(uintptr_t)(x + (size_t)blockIdx.x * BQ * NCH * NT);
    const unsigned ldsBase = (unsigned)(uintptr_t)(&lds[0]);

    // Scalar-provable leader gate (TDM ignores EXEC, so we need a real branch).
    const bool leader = (__builtin_amdgcn_readfirstlane((int)threadIdx.x) == 0);

    if (leader) tdm_issue(ldsBase, gbase);                // prime tile 0 -> buf 0

    for (int it = 0; it < NTILES; ++it) {
        if (leader) {
            if (it + 1 < NTILES) {
                // prefetch next tile into the other buffer, then wait for tile `it`
                tdm_issue(ldsBase + (unsigned)(((it + 1) & 1) * BUF_FLOATS * 4),
                          gbase + (unsigned long long)(it + 1) * TN * 4ull);
                __builtin_amdgcn_s_wait_tensorcnt(1);
            } else {
                __builtin_amdgcn_s_wait_tensorcnt(0);
            }
        }
        __syncthreads();   // tile `it` visible to all waves

        const float* Lb = &lds[(size_t)(it & 1) * BUF_FLOATS + (size_t)tid * (NCH * ROWF)];
        const int kmax = (it == NTILES - 1) ? (NT - it * TN) : TN;

        for (int k = 0; k < kmax; ++k) {
            // conflict-free LDS reads (thread stride 119 floats, gcd(119,64)=1)
            const float tc = Lb[k];
            const float r0 = Lb[1 * ROWF + k], r1 = Lb[2 * ROWF + k], r2 = Lb[3 * ROWF + k];
            const float r3 = Lb[4 * ROWF + k], r4 = Lb[5 * ROWF + k], r5 = Lb[6 * ROWF + k];

            // fused calibration: imu_c = (I + calib) * imu + bias
            const float g0c = M[0][0]*r0 + M[0][1]*r1 + M[0][2]*r2 + M[0][3]*r3 + M[0][4]*r4 + M[0][5]*r5 + bv[0];
            const float g1c = M[1][0]*r0 + M[1][1]*r1 + M[1][2]*r2 + M[1][3]*r3 + M[1][4]*r4 + M[1][5]*r5 + bv[1];
            const float g2c = M[2][0]*r0 + M[2][1]*r1 + M[2][2]*r2 + M[2][3]*r3 + M[2][4]*r4 + M[2][5]*r5 + bv[2];
            const float a0c = M[3][0]*r0 + M[3][1]*r1 + M[3][2]*r2 + M[3][3]*r3 + M[3][4]*r4 + M[3][5]*r5 + bv[3];
            const float a1c = M[4][0]*r0 + M[4][1]*r1 + M[4][2]*r2 + M[4][3]*r3 + M[4][4]*r4 + M[4][5]*r5 + bv[4];
            const float a2c = M[5][0]*r0 + M[5][1]*r1 + M[5][2]*r2 + M[5][3]*r3 + M[5][4]*r4 + M[5][5]*r5 + bv[5];

            if (have) {
                const float dt = tc - tP;

                // so3_exp(w*dt):  E = (1 - Bc*th2)*I + A*K + Bc*w w^T   (K^2 = w w^T - th2*I)
                const float wx = wP0 * dt, wy = wP1 * dt, wz = wP2 * dt;
                const float th2 = wx * wx + wy * wy + wz * wz;
                const float th  = sqrtf(th2);
                const bool  sm  = th < 1e-7f;
                const float safe = sm ? 1.0f : th;
                const float A  = sm ? 1.0f : (sinf(safe) / safe);
                const float Bc = sm ? 0.5f : ((1.0f - cosf(safe)) / (safe * safe));
                const float cI = 1.0f - Bc * th2;
                const float E00 = cI + Bc * wx * wx;
                const float E01 = Bc * wx * wy - A * wz;
                const float E02 = Bc * wx * wz + A * wy;
                const float E10 = Bc * wx * wy + A * wz;
                const float E11 = cI + Bc * wy * wy;
                const float E12 = Bc * wy * wz - A * wx;
                const float E20 = Bc * wx * wz - A * wy;
                const float E21 = Bc * wy * wz + A * wx;
                const float E22 = cI + Bc * wz * wz;

                // Ca = DC @ a
                const float Cax = DC00 * aP0 + DC01 * aP1 + DC02 * aP2;
                const float Cay = DC10 * aP0 + DC11 * aP1 + DC12 * aP2;
                const float Caz = DC20 * aP0 + DC21 * aP1 + DC22 * aP2;

                const float h = 0.5f * dt * dt;
                DR0 += DV0 * dt + Cax * h;
                DR1 += DV1 * dt + Cay * h;
                DR2 += DV2 * dt + Caz * h;
                DV0 += Cax * dt;
                DV1 += Cay * dt;
                DV2 += Caz * dt;

                // DC = DC @ E
                const float n00 = DC00 * E00 + DC01 * E10 + DC02 * E20;
                const float n01 = DC00 * E01 + DC01 * E11 + DC02 * E21;
                const float n02 = DC00 * E02 + DC01 * E12 + DC02 * E22;
                const float n10 = DC10 * E00 + DC11 * E10 + DC12 * E20;
                const float n11 = DC10 * E01 + DC11 * E11 + DC12 * E21;
                const float n12 = DC10 * E02 + DC11 * E12 + DC12 * E22;
                const float n20 = DC20 * E00 + DC21 * E10 + DC22 * E20;
                const float n21 = DC20 * E01 + DC21 * E11 + DC22 * E21;
                const float n22 = DC20 * E02 + DC21 * E12 + DC22 * E22;
                DC00 = n00; DC01 = n01; DC02 = n02;
                DC10 = n10; DC11 = n11; DC12 = n12;
                DC20 = n20; DC21 = n21; DC22 = n22;
            }
            have = true;
            tP = tc;
            wP0 = g0c; wP1 = g1c; wP2 = g2c;
            aP0 = a0c; aP1 = a1c; aP2 = a2c;
        }
        __syncthreads();   // done reading buf (it&1); next iteration may DMA into it
    }

    float* o = out + (size_t)b * 15;
    o[0]  = DR0;  o[1]  = DR1;  o[2]  = DR2;
    o[3]  = DV0;  o[4]  = DV1;  o[5]  = DV2;
    o[6]  = DC00; o[7]  = DC01; o[8]  = DC02;
    o[9]  = DC10; o[10] = DC11; o[11] = DC12;
    o[12] = DC20; o[13] = DC21; o[14] = DC22;
}

extern "C" void kernel_launch(void* const* d_in, const int* in_sizes, int n_in,
                              void* d_out, int out_size, void* d_ws, size_t ws_size,
                              hipStream_t stream) {
    const float* x     = (const float*)d_in[0];   // (32768, 7, 200)
    const float* calib = (const float*)d_in[1];   // (6, 6)
    const float* bias6 = (const float*)d_in[2];   // (6, 1)
    float*       out   = (float*)d_out;           // (32768, 15)
    (void)in_sizes; (void)n_in; (void)out_size; (void)d_ws; (void)ws_size;

    rmi_preint_kernel<<<BATCH / BQ, BQ, 0, stream>>>(x, calib, bias6, out);
}